// UpSamplingWithIndices_75771813036279
// MI455X (gfx1250) — compile-verified
//
#include <hip/hip_runtime.h>

// Max-unpooling scatter for B=32, H=W=64, C=128 (fixed by the reference).
//   pooled elems / batch   = H*W*C   = 2^19
//   unpooled elems / batch = 4*H*W*C = 2^21   (power of two -> mod becomes AND)
// out[b*2^21 + (am & (2^21-1))] += v,  b = flat_input_index >> 19.

typedef float f32x4 __attribute__((ext_vector_type(4)));
typedef int   i32x4 __attribute__((ext_vector_type(4)));

#define IMAGE_SHIFT 19
#define OUT_SHIFT   21
#define OUT_MASK    ((1u << OUT_SHIFT) - 1u)

// Pass 1: zero the 256MB output with b128 stores (RT hint: keep lines in the
// 192MB L2 so the scatter's atomics resolve in-cache).
__global__ void zero_f32x4(f32x4* __restrict__ out, int n4) {
  int i = blockIdx.x * blockDim.x + threadIdx.x;
  int stride = gridDim.x * blockDim.x;
  f32x4 z = (f32x4)0.0f;
  for (; i < n4; i += stride)
    out[i] = z;
}

// Pass 2: streaming scatter-add. Each thread handles 4 consecutive elements
// (one global_load_b128 of values + one of argmax), then 4 fire-and-forget
// global_atomic_add_f32. NT hints on the read-once input streams; prefetch
// the next grid-stride iteration.
__global__ void unpool_scatter(const f32x4* __restrict__ vals,
                               const i32x4* __restrict__ amax,
                               float* __restrict__ out, int ngroups) {
  int i = blockIdx.x * blockDim.x + threadIdx.x;
  int stride = gridDim.x * blockDim.x;
  for (; i < ngroups; i += stride) {
    if (i + stride < ngroups) {
      __builtin_prefetch(&vals[i + stride], 0, 0);   // -> global_prefetch_b8
      __builtin_prefetch(&amax[i + stride], 0, 0);
    }
    f32x4 v = __builtin_nontemporal_load(&vals[i]);  // NT: read-once stream
    i32x4 a = __builtin_nontemporal_load(&amax[i]);

    // 2^19 % 4 == 0, so all 4 elements of this group share one batch id.
    int b = (i << 2) >> IMAGE_SHIFT;
    float* outb = out + ((long long)b << OUT_SHIFT);

    unsafeAtomicAdd(outb + (((unsigned)a.x) & OUT_MASK), v.x);
    unsafeAtomicAdd(outb + (((unsigned)a.y) & OUT_MASK), v.y);
    unsafeAtomicAdd(outb + (((unsigned)a.z) & OUT_MASK), v.z);
    unsafeAtomicAdd(outb + (((unsigned)a.w) & OUT_MASK), v.w);
  }
}

extern "C" void kernel_launch(void* const* d_in, const int* in_sizes, int n_in,
                              void* d_out, int out_size, void* d_ws, size_t ws_size,
                              hipStream_t stream) {
  const float* vals = (const float*)d_in[0];   // [B,H,W,C] fp32, 16,777,216 elems
  const int*   amax = (const int*)d_in[1];     // [B,H,W,C] int32
  float*       out  = (float*)d_out;           // [B,2H,2W,C] fp32, 67,108,864 elems

  int n4_out  = out_size / 4;        // 16,777,216 float4 stores
  int ngroups = in_sizes[0] / 4;     // 4,194,304 groups of 4 input elems

  // 256 threads = 8 wave32s per block.
  zero_f32x4<<<8192, 256, 0, stream>>>((f32x4*)out, n4_out);
  unpool_scatter<<<4096, 256, 0, stream>>>((const f32x4*)vals,
                                           (const i32x4*)amax, out, ngroups);
}